// GraphEncoder_27204322853378
// MI455X (gfx1250) — compile-verified
//
#include <hip/hip_runtime.h>
#include <hip/hip_bf16.h>
#include <cstddef>

// ---------------------------------------------------------------------------
// GCNConv(50->128) + ReLU + mean-pool + Linear(128->64) + tanh  (MI455X/gfx1250)
// Dense GEMM on the fp32 WMMA pipe (V_WMMA_F32_16X16X4_F32, exact fp32),
// branchless inner loop. Edge scatter with native f32 L2 atomics; the 25.6MB
// h/out arrays are L2-resident (192MB L2), so the edge pass is L2-bound.
// ---------------------------------------------------------------------------

typedef __attribute__((ext_vector_type(2))) float v2f;
typedef __attribute__((ext_vector_type(8))) float v8f;

#define IN_FEATS 50
#define HID      128
#define OUTF     64

// ---------------- init: zero out-accumulator & pool, deg = 1.0 (self loop) --
__global__ void init_kernel(float* __restrict__ out_acc, float* __restrict__ deg,
                            float* __restrict__ pool, int n_nodes) {
    int idx = blockIdx.x * blockDim.x + threadIdx.x;
    int total = n_nodes * HID;
    if (idx < total) out_acc[idx] = 0.0f;
    if (idx < n_nodes) deg[idx] = 1.0f;      // self-loop contributes 1 to deg[dst]
    if (idx < HID) pool[idx] = 0.0f;
}

// ---------------- GEMM h = X @ W  via fp32 WMMA 16x16x4 ---------------------
// One wave computes one 16x16 tile of h. K=50: 12 unguarded steps + 1 tail.
// Row guard is address-clamp only: garbage A rows feed only unstored D rows.
__global__ void gemm_wmma_f32(const float* __restrict__ x,
                              const float* __restrict__ W,
                              float* __restrict__ h, int n_nodes) {
    int wid  = (blockIdx.x * blockDim.x + threadIdx.x) >> 5;
    int lane = threadIdx.x & 31;
    int row_tiles = (n_nodes + 15) >> 4;
    if (wid >= row_tiles * (HID / 16)) return;

    int ct = wid & 7;          // column tile (128/16 = 8)
    int rt = wid >> 3;         // row tile
    int m16 = lane & 15;
    int row = rt * 16 + m16;   // A row for this lane
    int col = ct * 16 + m16;   // B/D column for this lane
    int rowc = row < n_nodes ? row : n_nodes - 1;   // clamp for address safety
    int kh = (lane >> 4) << 1; // lanes 0-15 -> K+{0,1}; lanes 16-31 -> K+{2,3}
    bool hi = lane >= 16;

    const float2* xrow = (const float2*)(x + (size_t)rowc * IN_FEATS);

    v8f c = {};
    // 12 fully in-bounds K-steps (ka+1 <= 47 < 50): branchless loads
    #pragma unroll
    for (int k0 = 0; k0 < 48; k0 += 4) {
        int ka = k0 + kh;
        float2 av = xrow[ka >> 1];          // 8B-aligned: index even
        v2f a, b;
        a.x = av.x;
        a.y = av.y;
        b.x = W[(size_t)ka * HID + col];
        b.y = W[(size_t)(ka + 1) * HID + col];
        c = __builtin_amdgcn_wmma_f32_16x16x4_f32(false, a, false, b,
                                                  (short)0, c, false, false);
    }
    // tail K-step: lanes 0-15 cover K=48,49; lanes 16-31 would be K=50,51 -> 0
    {
        float2 av = xrow[24];               // x[row][48], x[row][49]
        float w0 = W[(size_t)48 * HID + col];
        float w1 = W[(size_t)49 * HID + col];
        v2f a, b;
        a.x = hi ? 0.0f : av.x;
        a.y = hi ? 0.0f : av.y;
        b.x = hi ? 0.0f : w0;
        b.y = hi ? 0.0f : w1;
        c = __builtin_amdgcn_wmma_f32_16x16x4_f32(false, a, false, b,
                                                  (short)0, c, false, false);
    }

    // C/D layout: VGPR v -> row (v + 0|8 by lane half), col = lane&15
    int rbase = rt * 16 + ((lane >> 4) << 3);
    #pragma unroll
    for (int v = 0; v < 8; ++v) {
        int r = rbase + v;
        if (r < n_nodes) h[(size_t)r * HID + col] = c[v];
    }
}

// ---------------- degree: deg[dst] += 1 per edge ----------------------------
__global__ void degree_kernel(const long long* __restrict__ ei, float* __restrict__ deg,
                              int n_edges) {
    int e = blockIdx.x * blockDim.x + threadIdx.x;
    if (e >= n_edges) return;
    long long d = ei[(size_t)n_edges + e];   // row 1 = dst
    unsafeAtomicAdd(&deg[d], 1.0f);
}

// ---------------- dinv = rsqrt(deg)  (deg >= 1 always) ----------------------
__global__ void rsqrt_kernel(const float* __restrict__ deg, float* __restrict__ dinv,
                             int n_nodes) {
    int i = blockIdx.x * blockDim.x + threadIdx.x;
    if (i >= n_nodes) return;
    dinv[i] = __frsqrt_rn(deg[i]);
}

// ---------------- edge scatter: out[dst] += h[src] * dinv[src]*dinv[dst] ----
// One wave per edge; each lane handles 4 channels (float4 gather, 4 atomics).
__global__ void scatter_kernel(const float* __restrict__ h,
                               const float* __restrict__ dinv,
                               const long long* __restrict__ ei,
                               float* __restrict__ out_acc, int n_edges) {
    int wave = (blockIdx.x * blockDim.x + threadIdx.x) >> 5;
    int lane = threadIdx.x & 31;
    if (wave >= n_edges) return;
    long long s = ei[wave];
    long long d = ei[(size_t)n_edges + wave];
    float norm = dinv[s] * dinv[d];
    const float4* hrow = (const float4*)(h + (size_t)s * HID);
    float4 v = hrow[lane];                       // channels lane*4 .. lane*4+3
    float* orow = out_acc + (size_t)d * HID + lane * 4;
    unsafeAtomicAdd(orow + 0, v.x * norm);
    unsafeAtomicAdd(orow + 1, v.y * norm);
    unsafeAtomicAdd(orow + 2, v.z * norm);
    unsafeAtomicAdd(orow + 3, v.w * norm);
}

// ---------------- pool: sum_i relu(out[i] + h[i]*dinv[i]^2 + b) per channel -
__global__ void pool_kernel(const float* __restrict__ out_acc,
                            const float* __restrict__ h,
                            const float* __restrict__ dinv,
                            const float* __restrict__ b_gcn,
                            float* __restrict__ pool, int n_nodes) {
    int c = threadIdx.x;                 // 0..127
    float bc = b_gcn[c];
    float acc = 0.0f;
    for (int i = blockIdx.x; i < n_nodes; i += gridDim.x) {
        float di = dinv[i];
        float v = out_acc[(size_t)i * HID + c]
                + h[(size_t)i * HID + c] * di * di    // self-loop message
                + bc;
        acc += fmaxf(v, 0.0f);
    }
    unsafeAtomicAdd(&pool[c], acc);
}

// ---------------- final: tanh(mean(h) @ W_lin + b_lin) ----------------------
__global__ void final_kernel(const float* __restrict__ pool,
                             const float* __restrict__ W_lin,
                             const float* __restrict__ b_lin,
                             float* __restrict__ y, float inv_n) {
    int j = threadIdx.x;                 // 0..63
    if (j >= OUTF) return;
    float s = b_lin[j];
    #pragma unroll 4
    for (int c = 0; c < HID; ++c)
        s += (pool[c] * inv_n) * W_lin[(size_t)c * OUTF + j];
    y[j] = tanhf(s);
}

// ---------------------------------------------------------------------------
extern "C" void kernel_launch(void* const* d_in, const int* in_sizes, int n_in,
                              void* d_out, int out_size, void* d_ws, size_t ws_size,
                              hipStream_t stream) {
    const float*     x     = (const float*)d_in[0];
    const float*     W_gcn = (const float*)d_in[1];
    const float*     b_gcn = (const float*)d_in[2];
    const float*     W_lin = (const float*)d_in[3];
    const float*     b_lin = (const float*)d_in[4];
    const long long* ei    = (const long long*)d_in[5];

    int n_nodes = in_sizes[0] / IN_FEATS;     // 50000
    int n_edges = in_sizes[5] / 2;            // 1,600,000

    // workspace layout (floats)
    float* ws      = (float*)d_ws;
    float* h       = ws;                                  // n_nodes*128
    float* out_acc = h + (size_t)n_nodes * HID;           // n_nodes*128
    float* deg     = out_acc + (size_t)n_nodes * HID;     // n_nodes
    float* dinv    = deg + n_nodes;                       // n_nodes
    float* pool    = dinv + n_nodes;                      // 128

    float* y = (float*)d_out;

    // 1) init accumulators
    {
        int total = n_nodes * HID;
        int blocks = (total + 255) / 256;
        init_kernel<<<blocks, 256, 0, stream>>>(out_acc, deg, pool, n_nodes);
    }
    // 2) h = X @ W_gcn   (fp32 WMMA)
    {
        int row_tiles = (n_nodes + 15) / 16;
        int waves = row_tiles * (HID / 16);
        int blocks = (waves * 32 + 255) / 256;
        gemm_wmma_f32<<<blocks, 256, 0, stream>>>(x, W_gcn, h, n_nodes);
    }
    // 3) degrees
    {
        int blocks = (n_edges + 255) / 256;
        degree_kernel<<<blocks, 256, 0, stream>>>(ei, deg, n_edges);
    }
    // 4) dinv
    {
        int blocks = (n_nodes + 255) / 256;
        rsqrt_kernel<<<blocks, 256, 0, stream>>>(deg, dinv, n_nodes);
    }
    // 5) edge scatter (one wave per edge)
    {
        int waves_per_block = 256 / 32;
        int blocks = (n_edges + waves_per_block - 1) / waves_per_block;
        scatter_kernel<<<blocks, 256, 0, stream>>>(h, dinv, ei, out_acc, n_edges);
    }
    // 6) column pool (ReLU + self-loop + bias folded in)
    {
        pool_kernel<<<256, HID, 0, stream>>>(out_acc, h, dinv, b_gcn, pool, n_nodes);
    }
    // 7) final linear + tanh
    {
        final_kernel<<<1, 64, 0, stream>>>(pool, W_lin, b_lin, y, 1.0f / (float)n_nodes);
    }
}